// chiral_multistack_86105504350332
// MI455X (gfx1250) — compile-verified
//
#include <hip/hip_runtime.h>
#include <cstdint>

#define WN 65536
#define LN 64
#define TPB 256
#define PIPE 5                 // chunks in flight (5-deep pipeline, 80KB LDS)
#define CHUNK 4                // layers per chunk (16B per array per lane)
#define NCHUNK (LN / CHUNK)    // 16

struct cx { float r, i; };

__device__ __forceinline__ cx cmul(cx a, cx b) {
  return { fmaf(a.r, b.r, -(a.i * b.i)), fmaf(a.r, b.i, a.i * b.r) };
}
__device__ __forceinline__ cx cadd(cx a, cx b) { return { a.r + b.r, a.i + b.i }; }

__device__ __forceinline__ float frcp(float x) {
  float r = __builtin_amdgcn_rcpf(x);
  return r * (2.0f - x * r);   // one Newton step
}

// CDNA5 async global->LDS copy, 16B per lane, tracked by ASYNCcnt.
__device__ __forceinline__ void async_ld16(const void* lds_p, const float* gp) {
  unsigned lo = (unsigned)(uintptr_t)lds_p;                 // flat LDS addr[31:0] == LDS offset
  unsigned long long ga = (unsigned long long)(uintptr_t)gp;
  asm volatile("global_load_async_to_lds_b128 %0, %1, off"
               :: "v"(lo), "v"(ga) : "memory");
}

// state' = state @ M_l where M = [[c, i s/Y],[i Y s, c]], delta = dr + i*di
__device__ __forceinline__ void layer_update(cx& A00, cx& A01, cx& A10, cx& A11,
                                             float dr, float di, cx n0, cx invY) {
  float sx, cz;
  __sincosf(dr, &sx, &cz);
  float e    = __expf(di);
  float einv = frcp(e);
  float ch = 0.5f * (e + einv);
  float sh = 0.5f * (e - einv);
  cx c  = { cz * ch, -sx * sh };        // cos(delta)
  cx s  = { sx * ch,  cz * sh };        // sin(delta)
  cx is = { -s.i, s.r };                // i*sin(delta)
  cx m01 = cmul(is, invY);
  cx m10 = cmul(is, n0);
  cx n00 = cadd(cmul(A00, c),   cmul(A01, m10));
  cx n01 = cadd(cmul(A00, m01), cmul(A01, c));
  cx n10 = cadd(cmul(A10, c),   cmul(A11, m10));
  cx n11 = cadd(cmul(A10, m01), cmul(A11, c));
  A00 = n00; A01 = n01; A10 = n10; A11 = n11;
}

__device__ __forceinline__ float absorbance(cx A00, cx A01, cx A10, cx A11) {
  const float NOUT = 1.45f;
  cx den = { A00.r + NOUT * A01.r + A10.r + NOUT * A11.r,
             A00.i + NOUT * A01.i + A10.i + NOUT * A11.i };
  float d2 = fmaf(den.r, den.r, den.i * den.i);
  float T  = NOUT * 4.0f * frcp(d2);            // N_OUT/N_IN * |2/den|^2 (N_IN=1)
  return -__logf(T + 1e-12f) * 0.43429448190325176f;  // -log10
}

__global__ void __launch_bounds__(TPB)
tmm_kernel(const float* __restrict__ wl,  const float* __restrict__ thickP,
           const float* __restrict__ mindp, const float* __restrict__ maxdp,
           const float* __restrict__ ecr, const float* __restrict__ eci,
           const float* __restrict__ kcr, const float* __restrict__ kci,
           const float* __restrict__ ear, const float* __restrict__ eai,
           const float* __restrict__ kar, const float* __restrict__ kai,
           float* __restrict__ out) {
  __shared__ float  ds_d[LN];
  __shared__ float4 stage[PIPE][4][TPB];   // 80 KB: [buf][array][lane], 16B slots

  const int tid = threadIdx.x;
  const int bps = WN / TPB;                   // blocks per stack
  const int s   = blockIdx.x / bps;           // 0 = stack c, 1 = stack a
  const int wb  = blockIdx.x - s * bps;
  const int w   = wb * TPB + tid;

  if (tid < LN) {
    float t  = thickP[tid];
    float sg = frcp(1.0f + __expf(-t));       // sigmoid
    ds_d[tid] = (sg * (maxdp[0] - mindp[0]) + mindp[0]) * 1e-9f;
  }
  __syncthreads();

  const float* er = s ? ear : ecr;
  const float* ei = s ? eai : eci;
  const float* kr = s ? kar : kcr;
  const float* ki = s ? kai : kci;

  const float k0   = 6.2831852e6f / wl[w];    // 2*pi / (wl*1e-6)
  const int   base = w * LN;

  auto issue = [&](int chunkIdx, int buf) {
    const int off = base + chunkIdx * CHUNK;
    async_ld16(&stage[buf][0][tid], er + off);
    async_ld16(&stage[buf][1][tid], ei + off);
    async_ld16(&stage[buf][2][tid], kr + off);
    async_ld16(&stage[buf][3][tid], ki + off);
  };

  // prologue: fill the 5-deep pipeline
#pragma unroll
  for (int c = 0; c < PIPE; ++c) issue(c, c);

  cx P00{1,0}, P01{0,0}, P10{0,0}, P11{1,0};  // sigma = +1
  cx Q00{1,0}, Q01{0,0}, Q10{0,0}, Q11{1,0};  // sigma = -1

  for (int i = 0; i < NCHUNK; ++i) {
    // ASYNCcnt retires in order: wait until chunk i's 4 loads are done.
    const int rem = NCHUNK - 1 - i;           // issued chunks still in flight beyond i
    if (rem >= 4)      asm volatile("s_wait_asynccnt 0x10" ::: "memory");
    else if (rem == 3) asm volatile("s_wait_asynccnt 0xc"  ::: "memory");
    else if (rem == 2) asm volatile("s_wait_asynccnt 0x8"  ::: "memory");
    else if (rem == 1) asm volatile("s_wait_asynccnt 0x4"  ::: "memory");
    else               asm volatile("s_wait_asynccnt 0x0"  ::: "memory");

    const int buf = i % PIPE;
    float4 ve  = stage[buf][0][tid];
    float4 vei = stage[buf][1][tid];
    float4 vkr = stage[buf][2][tid];
    float4 vki = stage[buf][3][tid];
    float4 dq  = *(const float4*)&ds_d[i * CHUNK];

    // make sure the LDS reads of this buffer completed before we let the
    // async engine overwrite it with chunk i+PIPE
    asm volatile("s_wait_dscnt 0x0" ::: "memory");
    const int nx = i + PIPE;
    if (nx < NCHUNK) issue(nx, buf);

    float aer[CHUNK] = { ve.x,  ve.y,  ve.z,  ve.w  };
    float aei[CHUNK] = { vei.x, vei.y, vei.z, vei.w };
    float akr[CHUNK] = { vkr.x, vkr.y, vkr.z, vkr.w };
    float aki[CHUNK] = { vki.x, vki.y, vki.z, vki.w };
    float add[CHUNK] = { dq.x,  dq.y,  dq.z,  dq.w  };

#pragma unroll
    for (int j = 0; j < CHUNK; ++j) {
      float epr = aer[j], epi = aei[j];
      float kpr = akr[j], kpi = aki[j];
      // n0 = sqrt(eps), eps_re > 0, eps_im >= 0
      float mag = __builtin_amdgcn_sqrtf(fmaf(epr, epr, epi * epi));
      float n0r = __builtin_amdgcn_sqrtf(0.5f * (mag + epr));
      float n0i = epi * 0.5f * frcp(n0r);
      cx n0 = { n0r, n0i };
      float inv2 = frcp(fmaf(n0r, n0r, n0i * n0i));
      cx invY = { n0r * inv2, -n0i * inv2 };
      float kd  = k0 * add[j];
      float ar_ = kd * n0r, ai_ = kd * n0i;   // kd * n0
      float br_ = kd * kpr, bi_ = kd * kpi;   // kd * kappa
      layer_update(P00, P01, P10, P11, ar_ + br_, ai_ + bi_, n0, invY); // sigma=+1
      layer_update(Q00, Q01, Q10, Q11, ar_ - br_, ai_ - bi_, n0, invY); // sigma=-1
    }
  }

  float A_r = absorbance(P00, P01, P10, P11);  // sigma = +1
  float A_l = absorbance(Q00, Q01, Q10, Q11);  // sigma = -1
  const float SCALE = (2.0f * 0.87f - 1.0f) * 32980.0f;
  out[s * WN + w]       = (A_l - A_r) * SCALE;   // cd_c / cd_a
  out[(2 + s) * WN + w] = 0.5f * (A_l + A_r);    // abs_c / abs_a
}

extern "C" void kernel_launch(void* const* d_in, const int* in_sizes, int n_in,
                              void* d_out, int out_size, void* d_ws, size_t ws_size,
                              hipStream_t stream) {
  (void)in_sizes; (void)n_in; (void)out_size; (void)d_ws; (void)ws_size;
  const float* wl     = (const float*)d_in[0];
  const float* thickP = (const float*)d_in[1];
  const float* mind   = (const float*)d_in[2];
  const float* maxd   = (const float*)d_in[3];
  const float* ecr    = (const float*)d_in[4];
  const float* eci    = (const float*)d_in[5];
  const float* kcr    = (const float*)d_in[6];
  const float* kci    = (const float*)d_in[7];
  const float* ear    = (const float*)d_in[8];
  const float* eai    = (const float*)d_in[9];
  const float* kar    = (const float*)d_in[10];
  const float* kai    = (const float*)d_in[11];

  dim3 grid(2 * (WN / TPB));   // 512 blocks: first 256 = stack c, next 256 = stack a
  dim3 block(TPB);
  tmm_kernel<<<grid, block, 0, stream>>>(wl, thickP, mind, maxd,
                                         ecr, eci, kcr, kci,
                                         ear, eai, kar, kai,
                                         (float*)d_out);
}